// GAT_66288525247250
// MI455X (gfx1250) — compile-verified
//
#include <hip/hip_runtime.h>
#include <hip/hip_bf16.h>
#include <math.h>
#include <stdint.h>

typedef __attribute__((ext_vector_type(2))) float    v2f;
typedef __attribute__((ext_vector_type(8))) float    v8f;
typedef __attribute__((ext_vector_type(4))) unsigned u32x4;
typedef __attribute__((ext_vector_type(8))) int      i32x8;
typedef __attribute__((ext_vector_type(4))) int      i32x4;

#define NEG_SLOPE 0.2f

static constexpr int N_NODES = 50000;
static constexpr int N_EDGES = 800000;
static constexpr int ETOT    = N_EDGES + N_NODES;   // self-loops appended implicitly
static constexpr int F_IN    = 256;
static constexpr int H1 = 4, C1 = 64, F1 = 256;     // H1*C1
static constexpr int H2 = 1, C2 = 64, F2 = 64;

static constexpr int KMAX      = 256;   // K is 256 for every GEMM here
static constexpr int BS_STRIDE = 68;    // padded LDS row stride (bank-conflict free, 16B-aligned)

// ---- monotone float <-> ordered-uint key (for atomicMax on floats) ----
__device__ __forceinline__ unsigned f2key(float f) {
  unsigned u = __float_as_uint(f);
  return (u & 0x80000000u) ? ~u : (u | 0x80000000u);
}
__device__ __forceinline__ float key2f(unsigned k) {
  unsigned u = (k & 0x80000000u) ? (k & 0x7fffffffu) : ~k;
  return __uint_as_float(u);
}

// ================= WMMA GEMM: C = act(A @ B + bias) =================
// A: MxK row-major, B: KxN row-major, C: MxN row-major. 8 <= K <= 256, K%8==0,
// M%16==0, N%64==0. Grid: x = ceil(M/128) (8 waves x 16 rows), y = N/64.
// Block stages the K x 64 B-panel in LDS (TDM tensor_load_to_lds with hardware
// row padding when available, cooperative float4 fill otherwise); each wave
// computes a 16x64 tile with a ping-pong software-pipelined K-loop.
// act: 0 = none, 1 = relu

#define LOAD_STAGE(ax, bxx, byy)                                                \
  do {                                                                          \
    ax = *(const v2f*)ak; ak += 4;                                              \
    _Pragma("unroll")                                                           \
    for (int t = 0; t < 4; ++t) {                                               \
      bxx[t] = bp[t * 16];                                                      \
      byy[t] = bp[BS_STRIDE + t * 16];                                          \
    }                                                                           \
    bp += 4 * BS_STRIDE;                                                        \
  } while (0)

#define COMPUTE_STAGE(ax, bxx, byy)                                             \
  do {                                                                          \
    _Pragma("unroll")                                                           \
    for (int t = 0; t < 4; ++t) {                                               \
      v2f b; b.x = bxx[t]; b.y = byy[t];                                        \
      acc[t] = __builtin_amdgcn_wmma_f32_16x16x4_f32(                           \
          false, ax, false, b, (short)0, acc[t], false, false);                 \
    }                                                                           \
  } while (0)

__global__ __launch_bounds__(256)
void gemm_wmma_kernel(const float* __restrict__ A,
                      const float* __restrict__ B,
                      const float* __restrict__ bias,
                      float* __restrict__ C,
                      int M, int K, int N, int act)
{
  __shared__ float Bs[KMAX * BS_STRIDE];

  const int tid = (int)threadIdx.x;
  const int n0  = (int)blockIdx.y << 6;

#if __has_builtin(__builtin_amdgcn_tensor_load_to_lds)
  // ---- Tensor Data Mover: DMA the K x 64 panel into LDS with row padding ----
  // LDS row stride = 64 data DWORDs + 4 pad DWORDs = BS_STRIDE floats.
  if (tid < 32) {                                  // one wave issues the TDM op
    const unsigned           ldsAddr = (unsigned)(uintptr_t)(void*)Bs;  // low 32b = LDS offset
    const unsigned long long gaddr   = (unsigned long long)(uintptr_t)(B + n0);
    u32x4 g0;
    g0[0] = 1u;                                            // count=1 (valid user D#)
    g0[1] = ldsAddr;                                       // lds_addr
    g0[2] = (unsigned)(gaddr & 0xffffffffull);             // global_addr[31:0]
    g0[3] = (unsigned)((gaddr >> 32) & 0x01ffffffull)      // global_addr[56:32]
            | (2u << 30);                                  // type = 2 ("image")
    i32x8 g1;
    g1[0] = (2 << 16)        // data_size = 4B
          | (1 << 20)        // pad_enable
          | (5 << 22)        // pad_interval: 64 DWORDs
          | (3 << 25);       // pad_amount: 4 DWORDs
    g1[1] = (int)(((unsigned)N & 0xffffu) << 16);                    // tensor_dim0[15:0]
    g1[2] = (int)(((unsigned)N >> 16) | (((unsigned)K & 0xffffu) << 16)); // dim0 hi | dim1 lo
    g1[3] = (int)(((unsigned)K >> 16) | (64u << 16));                // dim1 hi | tile_dim0=64
    g1[4] = (int)(unsigned)K;                                        // tile_dim1=K, tile_dim2=0
    g1[5] = (int)(unsigned)N;                                        // tensor_dim0_stride = N
    g1[6] = 0;
    g1[7] = 0;
    i32x4 gz4 = {0, 0, 0, 0};                              // groups 2/3 unused (2D tensor)
    i32x8 gz8 = {0, 0, 0, 0, 0, 0, 0, 0};
    __builtin_amdgcn_tensor_load_to_lds(g0, g1, gz4, gz4, gz8, 0);
    __builtin_amdgcn_s_wait_tensorcnt(0);
  }
  __syncthreads();
#else
  // ---- cooperative fill of the K x 64 panel (float4 chunks) ----
  for (int i = tid; i < (K << 4); i += 256) {        // K*16 float4 chunks
    const int row = i >> 4;
    const int c4  = (i & 15) << 2;
    const float4 v = *(const float4*)(B + (size_t)row * N + n0 + c4);
    *(float4*)(&Bs[row * BS_STRIDE + c4]) = v;
  }
  __syncthreads();
#endif

  const int wv = tid >> 5;
  const int m0 = (((int)blockIdx.x << 3) + wv) << 4;
  if (m0 >= M) return;                    // wave-uniform: EXEC all-1s at WMMA

  const int lane = tid & 31;
  const int l    = lane & 15;
  const int half = lane >> 4;             // lanes 16-31 carry K+2/K+3

  const float* ak = A + (size_t)(m0 + l) * K + (half << 1);   // global, streamed
  const float* bp = Bs + (half << 1) * BS_STRIDE + l;         // LDS

  v8f acc[4] = {};

  // ---- ping-pong pipelined K loop (two stages, no register rotation) ----
  v2f a0, a1;
  float bx0[4], by0[4], bx1[4], by1[4];
  LOAD_STAGE(a0, bx0, by0);               // k
  LOAD_STAGE(a1, bx1, by1);               // k+4
  for (int k = 0; k < K - 8; k += 8) {
    COMPUTE_STAGE(a0, bx0, by0);
    LOAD_STAGE(a0, bx0, by0);             // k+8
    COMPUTE_STAGE(a1, bx1, by1);
    LOAD_STAGE(a1, bx1, by1);             // k+12
  }
  COMPUTE_STAGE(a0, bx0, by0);            // K-8
  COMPUTE_STAGE(a1, bx1, by1);            // K-4

  // C/D layout: VGPR r -> row m0 + r + 8*half, col tile_n0 + (lane&15)
  const int rbase = m0 + (half << 3);
#pragma unroll
  for (int t = 0; t < 4; ++t) {
    const int col = n0 + t * 16 + l;
    const float bv = bias ? bias[col] : 0.0f;
#pragma unroll
    for (int r = 0; r < 8; ++r) {
      float v = acc[t][r] + bv;
      if (act == 1) v = fmaxf(v, 0.0f);
      C[(size_t)(rbase + r) * N + col] = v;
    }
  }
}

// ============ per-node attention logits: al = xp . a  (per head) ============
template <int H, int C>
__global__ __launch_bounds__(256)
void al_kernel(const float* __restrict__ xp,
               const float* __restrict__ a_src,
               const float* __restrict__ a_dst,
               float* __restrict__ alS, float* __restrict__ alD, int Nn)
{
  int i = (int)(blockIdx.x * blockDim.x + threadIdx.x);
  if (i >= Nn * H) return;
  int n = i / H, h = i % H;
  const float* xr = xp + (size_t)n * (H * C) + h * C;
  const float* as = a_src + h * C;
  const float* ad = a_dst + h * C;
  float s = 0.f, d = 0.f;
#pragma unroll 8
  for (int c = 0; c < C; ++c) {
    float x = xr[c];
    s = fmaf(x, as[c], s);
    d = fmaf(x, ad[c], d);
  }
  alS[i] = s;
  alD[i] = d;
}

// ======================= zero fill (capture-safe memset) =======================
__global__ __launch_bounds__(256)
void zero_kernel(float* __restrict__ p, int n)
{
  int i = (int)(blockIdx.x * blockDim.x + threadIdx.x);
  if (i < n) p[i] = 0.0f;
}

// ============== edge pass 1: segment max of leaky_relu(alS+alD) ==============
template <int H>
__global__ __launch_bounds__(256)
void edge_max_kernel(const int* __restrict__ src, const int* __restrict__ dst,
                     int E, int Nn,
                     const float* __restrict__ alS, const float* __restrict__ alD,
                     unsigned* __restrict__ amaxKey)
{
  int e = (int)(blockIdx.x * blockDim.x + threadIdx.x);
  if (e >= E + Nn) return;
  int s, d;
  if (e < E) { s = src[e]; d = dst[e]; } else { s = d = e - E; }   // self-loops
#pragma unroll
  for (int h = 0; h < H; ++h) {
    float a = alS[s * H + h] + alD[d * H + h];
    a = (a > 0.f) ? a : NEG_SLOPE * a;
    atomicMax(&amaxKey[d * H + h], f2key(a));
  }
}

// ============== edge pass 2: denom = segment sum of exp(alpha-max) ==============
template <int H>
__global__ __launch_bounds__(256)
void edge_denom_kernel(const int* __restrict__ src, const int* __restrict__ dst,
                       int E, int Nn,
                       const float* __restrict__ alS, const float* __restrict__ alD,
                       const unsigned* __restrict__ amaxKey,
                       float* __restrict__ denom)
{
  int e = (int)(blockIdx.x * blockDim.x + threadIdx.x);
  if (e >= E + Nn) return;
  int s, d;
  if (e < E) { s = src[e]; d = dst[e]; } else { s = d = e - E; }
#pragma unroll
  for (int h = 0; h < H; ++h) {
    float a = alS[s * H + h] + alD[d * H + h];
    a = (a > 0.f) ? a : NEG_SLOPE * a;
    float ex = __expf(a - key2f(amaxKey[d * H + h]));
    atomicAdd(&denom[d * H + h], ex);
  }
}

// == edge pass 3: agg[dst] += attn * xp[src]  (one wave / edge; atomic f32 adds) ==
template <int H, int C>
__global__ __launch_bounds__(256)
void edge_agg_kernel(const int* __restrict__ src, const int* __restrict__ dst,
                     int E, int Nn,
                     const float* __restrict__ alS, const float* __restrict__ alD,
                     const unsigned* __restrict__ amaxKey,
                     const float* __restrict__ denom,
                     const float* __restrict__ xp,
                     float* __restrict__ agg)
{
  int e = (int)(blockIdx.x * (blockDim.x >> 5)) + (int)(threadIdx.x >> 5);
  if (e >= E + Nn) return;
  const int lane = (int)(threadIdx.x & 31);
  int s, d;
  if (e < E) { s = src[e]; d = dst[e]; } else { s = d = e - E; }

  // per-head attention weight, computed once per lane (registers)
  float w[H];
#pragma unroll
  for (int h = 0; h < H; ++h) {
    float a = alS[s * H + h] + alD[d * H + h];
    a = (a > 0.f) ? a : NEG_SLOPE * a;
    w[h] = __expf(a - key2f(amaxKey[d * H + h])) / denom[d * H + h];
  }

  constexpr int F = H * C;
  const float* xr = xp + (size_t)s * F;
  float* ar = agg + (size_t)d * F;
#pragma unroll
  for (int col = lane; col < F; col += 32) {
    atomicAdd(&ar[col], w[col / C] * xr[col]);
  }
}

// ===================== in-place bias + ELU on layer-1 output =====================
__global__ __launch_bounds__(256)
void elu_bias_kernel(float* __restrict__ p, const float* __restrict__ bias,
                     int n, int mask)
{
  int i = (int)(blockIdx.x * blockDim.x + threadIdx.x);
  if (i >= n) return;
  float v = p[i] + bias[i & mask];
  p[i] = (v > 0.f) ? v : (__expf(v) - 1.0f);
}

// ============================ d_out[i] += b2[i % F2] ============================
__global__ __launch_bounds__(256)
void add_bias_kernel(float* __restrict__ p, const float* __restrict__ bias,
                     int n, int mask)
{
  int i = (int)(blockIdx.x * blockDim.x + threadIdx.x);
  if (i < n) p[i] += bias[i & mask];
}

extern "C" void kernel_launch(void* const* d_in, const int* in_sizes, int n_in,
                              void* d_out, int out_size, void* d_ws, size_t ws_size,
                              hipStream_t stream) {
  const float* x_clean  = (const float*)d_in[0];
  const float* x_noised = (const float*)d_in[1];
  const int*   esrc     = (const int*)d_in[2];
  const int*   edst     = (const int*)d_in[3];
  const float* W1       = (const float*)d_in[4];
  const float* a_src1   = (const float*)d_in[5];
  const float* a_dst1   = (const float*)d_in[6];
  const float* b1       = (const float*)d_in[7];
  const float* W2       = (const float*)d_in[8];
  const float* a_src2   = (const float*)d_in[9];
  const float* a_dst2   = (const float*)d_in[10];
  const float* b2       = (const float*)d_in[11];
  float* out = (float*)d_out;

  // -------- workspace layout (floats); aggressive aliasing --------
  float* ws = (float*)d_ws;
  float*    XP1   = ws;                         // 12,800,000  (later reused as XP2)
  float*    HS    = ws + 12800000;              // 12,800,000  (later reused as AGG1 / h)
  float*    ALS1  = ws + 25600000;              //    200,000
  float*    ALD1  = ws + 25800000;              //    200,000
  unsigned* AMAX1 = (unsigned*)(ws + 26000000); //    200,000
  float*    DEN1  = ws + 26200000;              //    200,000
  float*    ALS2  = ws + 26400000;              //     50,000
  float*    ALD2  = ws + 26450000;              //     50,000
  unsigned* AMAX2 = (unsigned*)(ws + 26500000); //     50,000
  float*    DEN2  = ws + 26550000;              //     50,000
  float*    AGG1  = HS;                         // alias (hs dead after skip GEMM)
  float*    XP2   = ws;                         // alias (xp1 dead after layer-1 agg)

  const dim3 blk(256);
  auto cdiv = [](long long a, long long b) { return (unsigned)((a + b - 1) / b); };

  // ---- GEMMs (WMMA). grid: x = ceil((M/16)/8), y = N/64 ----
  const unsigned mBlocks = cdiv(N_NODES / 16, 8);  // 391
  const dim3 g1(mBlocks, F1 / 64);                 // (391, 4)
  const dim3 g2(mBlocks, F2 / 64);                 // (391, 1)

  // 1) xp1 = x_noised @ W1
  gemm_wmma_kernel<<<g1, blk, 0, stream>>>(x_noised, W1, nullptr, XP1,
                                           N_NODES, F_IN, F1, 0);
  // 2) hs = relu(x_clean @ W1 + b1)
  gemm_wmma_kernel<<<g1, blk, 0, stream>>>(x_clean, W1, b1, HS,
                                           N_NODES, F_IN, F1, 1);
  // 3) d_out = hs @ W2 + b2   (skip path, accumulated into later)
  gemm_wmma_kernel<<<g2, blk, 0, stream>>>(HS, W2, b2, out,
                                           N_NODES, F1, F2, 0);

  // ---- layer 1 attention ----
  al_kernel<H1, C1><<<cdiv(N_NODES * H1, 256), blk, 0, stream>>>(XP1, a_src1, a_dst1,
                                                                 ALS1, ALD1, N_NODES);
  zero_kernel<<<cdiv(400000, 256), blk, 0, stream>>>((float*)AMAX1, 400000);      // amax1+den1
  zero_kernel<<<cdiv(12800000, 256), blk, 0, stream>>>(AGG1, 12800000);           // agg1 (=HS)

  edge_max_kernel<H1><<<cdiv(ETOT, 256), blk, 0, stream>>>(esrc, edst, N_EDGES, N_NODES,
                                                           ALS1, ALD1, AMAX1);
  edge_denom_kernel<H1><<<cdiv(ETOT, 256), blk, 0, stream>>>(esrc, edst, N_EDGES, N_NODES,
                                                             ALS1, ALD1, AMAX1, DEN1);
  edge_agg_kernel<H1, C1><<<cdiv(ETOT, 8), blk, 0, stream>>>(esrc, edst, N_EDGES, N_NODES,
                                                             ALS1, ALD1, AMAX1, DEN1,
                                                             XP1, AGG1);
  // h = elu(agg1 + b1), in place
  elu_bias_kernel<<<cdiv(12800000, 256), blk, 0, stream>>>(AGG1, b1, 12800000, 255);

  // ---- layer 2 ----
  // xp2 = h @ W2 (reuses XP1 region)
  gemm_wmma_kernel<<<g2, blk, 0, stream>>>(AGG1, W2, nullptr, XP2,
                                           N_NODES, F1, F2, 0);
  al_kernel<H2, C2><<<cdiv(N_NODES * H2, 256), blk, 0, stream>>>(XP2, a_src2, a_dst2,
                                                                 ALS2, ALD2, N_NODES);
  zero_kernel<<<cdiv(100000, 256), blk, 0, stream>>>((float*)AMAX2, 100000);      // amax2+den2

  edge_max_kernel<H2><<<cdiv(ETOT, 256), blk, 0, stream>>>(esrc, edst, N_EDGES, N_NODES,
                                                           ALS2, ALD2, AMAX2);
  edge_denom_kernel<H2><<<cdiv(ETOT, 256), blk, 0, stream>>>(esrc, edst, N_EDGES, N_NODES,
                                                             ALS2, ALD2, AMAX2, DEN2);
  // aggregate layer-2 messages straight into d_out (which holds skip = hs@W2+b2)
  edge_agg_kernel<H2, C2><<<cdiv(ETOT, 8), blk, 0, stream>>>(esrc, edst, N_EDGES, N_NODES,
                                                             ALS2, ALD2, AMAX2, DEN2,
                                                             XP2, out);
  // final + b2 from the GAT layer itself
  add_bias_kernel<<<cdiv(N_NODES * F2, 256), blk, 0, stream>>>(out, b2, N_NODES * F2, 63);
}